// TimeDomainModel_20555713478790
// MI455X (gfx1250) — compile-verified
//
#include <hip/hip_runtime.h>
#include <hip/hip_bf16.h>

// Balloon-Windkessel BOLD, explicit Euler. T=1000 sequential steps, B=16384
// independent sims (one per lane). Noise is streamed [T,B] via the CDNA5
// Tensor Data Mover into triple-buffered LDS, overlapped with the serial
// per-sim dependence chain using TENSORcnt waits.

#define T_STEPS 1000
#define B_SIMS  16384
#define BLOCK   256
#define CT      20                    // time steps per TDM chunk
#define NCHUNK  (T_STEPS / CT)        // 50
#define NBUF    3                     // triple buffer: depth-2 prefetch

typedef __attribute__((ext_vector_type(4))) unsigned int u32x4;
typedef __attribute__((ext_vector_type(8))) int          i32x8;
typedef __attribute__((ext_vector_type(4))) int          i32x4;

// Issue one TDM 2D tile load: tile_w elements x tile_h rows, row stride
// row_stride elements, 4-byte elements, global byte address -> LDS byte addr.
__device__ __forceinline__ void tdm_load_tile_2d(unsigned lds_byte_addr,
                                                 unsigned long long gaddr,
                                                 unsigned tile_w,
                                                 unsigned tile_h,
                                                 unsigned long long row_stride) {
  // ---- D# group 0 (128b): count=1, lds_addr, global_addr[56:0], type=2
  u32x4 g0;
  g0.x = 1u;                                        // count=1 (valid), user mode
  g0.y = lds_byte_addr;                             // lds_addr (bytes)
  g0.z = (unsigned)(gaddr & 0xFFFFFFFFull);         // global_addr[31:0]
  g0.w = (unsigned)((gaddr >> 32) & 0x1FFFFFFull)   // global_addr[56:32]
       | (2u << 30);                                // type = 2 ("image")

  // ---- D# group 1 (256b)
  const unsigned td0 = 0x7FFFFFFFu;                 // tensor_dim0: huge -> no OOB zeroing
  const unsigned td1 = 0x7FFFFFFFu;                 // tensor_dim1
  const unsigned long long st0 = row_stride;        // tensor_dim0_stride (48b)
  i32x8 g1;
  g1[0] = (int)(2u << 16);                          // wg_mask=0; data_size=2 (4B); no pad/iter
  g1[1] = (int)((td0 & 0xFFFFu) << 16);             // abar_addr=0; td0[15:0]
  g1[2] = (int)(((td0 >> 16) & 0xFFFFu) | ((td1 & 0xFFFFu) << 16));
  g1[3] = (int)(((td1 >> 16) & 0xFFFFu) | ((tile_w & 0xFFFFu) << 16)); // tile_dim0
  g1[4] = (int)(tile_h & 0xFFFFu);                  // tile_dim1; tile_dim2=0
  g1[5] = (int)(st0 & 0xFFFFFFFFull);               // stride0[31:0]
  g1[6] = (int)((st0 >> 32) & 0xFFFFull);           // stride0[47:32]; stride1[15:0]=0
  g1[7] = 0;                                        // stride1[47:16]=0

  // ---- groups 2/3: unused dims (tile_dim3/4 = 0)
  i32x4 g2 = {0, 0, 0, 0};
  i32x4 g3 = {0, 0, 0, 0};
  // ---- arg4: no D# field (ISA VADDR4 is unused/NULL); zero-filled
  i32x8 g4 = {0, 0, 0, 0, 0, 0, 0, 0};

  __builtin_amdgcn_tensor_load_to_lds(g0, g1, g2, g3, g4, /*cpol=*/0);
}

__global__ __launch_bounds__(BLOCK)
void balloon_windkessel_kernel(const float* __restrict__ noise,
                               const float* __restrict__ p_sigma,
                               const float* __restrict__ p_mu,
                               const float* __restrict__ p_lamb,
                               const float* __restrict__ p_beta,
                               const float* __restrict__ p_psi,
                               const float* __restrict__ p_phi,
                               const float* __restrict__ p_chi,
                               float* __restrict__ out) {
  __shared__ float zbuf[NBUF][CT * BLOCK];          // 3 * 20KB = 60KB LDS

  const int tid = threadIdx.x;
  const int blk = blockIdx.x;
  const int b   = blk * BLOCK + tid;                // sim index

  // Uniform scalar params (single-element device arrays).
  const float sigma = *p_sigma;
  const float mu    = *p_mu;
  const float lamb  = *p_lamb;
  const float beta  = *p_beta;
  const float psi   = *p_psi;
  const float phi   = *p_phi;
  const float chi   = *p_chi;

  const float dt    = 0.01f;                        // DT
  const float v0c   = 0.02f;                        // V0
  const float namp  = 0.01f;                        // NOISE_AMP
  const float dt_l  = dt / lamb;
  const float rbeta = 1.0f / beta;
  // E(f) = 1 - (1-beta)^(1/f) = 1 - exp2(log2(1-beta)/f)
  const float l2    = __builtin_amdgcn_logf(1.0f - beta);  // v_log_f32 = log2

  // ALPHA == 1  =>  v^(1/alpha) == v  and  v_a*q/v == q.
  float s = 0.0f, f = 1.0f, v = 1.0f, q = 1.0f;

  const unsigned lds_base =
      (unsigned)(unsigned long long)(uintptr_t)(&zbuf[0][0]); // WG-relative LDS offset
  const unsigned long long gbase =
      (unsigned long long)(uintptr_t)(noise + (size_t)blk * BLOCK);
  const unsigned long long chunk_bytes = (unsigned long long)CT * B_SIMS * 4ull;
  const unsigned buf_bytes = CT * BLOCK * 4u;

  // Prime the pipeline: chunks 0 and 1 in flight, wait for chunk 0.
  if (tid == 0) {
    tdm_load_tile_2d(lds_base,             gbase,               BLOCK, CT, B_SIMS);
    tdm_load_tile_2d(lds_base + buf_bytes, gbase + chunk_bytes, BLOCK, CT, B_SIMS);
    __builtin_amdgcn_s_wait_tensorcnt(1);
  }
  __syncthreads();

  for (int k = 0; k < NCHUNK; ++k) {
    const float* zr = &zbuf[k % NBUF][tid];
    float* orow = out + (size_t)k * CT * B_SIMS + b;

    #pragma unroll 4
    for (int r = 0; r < CT; ++r) {
      const float z = zr[r * BLOCK];                         // ds_load_b32, conflict-free

      const float inv_f = __builtin_amdgcn_rcpf(f);          // v_rcp_f32
      const float E     = 1.0f - __builtin_amdgcn_exp2f(l2 * inv_f); // v_exp_f32

      const float s2 = s + dt * (namp * z - sigma * s - mu * (f - 1.0f));
      const float f2 = f + dt * s;
      const float v2 = v + dt_l * (f - v);                   // v_a == v (alpha=1)
      const float q2 = q + dt_l * (f * E * rbeta - q);       // v_a*q/v == q

      const float inv_v2 = __builtin_amdgcn_rcpf(v2);
      const float y = v0c * (phi * (1.0f - q2)
                           + psi * (1.0f - q2 * inv_v2)
                           + chi * (1.0f - v2));
      orow[(size_t)r * B_SIMS] = y;                          // coalesced b32 store

      s = s2; f = f2; v = v2; q = q2;
    }

    // Keep two TDM loads in flight; release buffer (k+2)%3 for refill
    // (all waves finished reading it in chunk k-1, guaranteed by the
    // barrier at the end of chunk k-1).
    if (tid == 0) {
      if (k + 2 < NCHUNK) {
        tdm_load_tile_2d(lds_base + (unsigned)((k + 2) % NBUF) * buf_bytes,
                         gbase + (unsigned long long)(k + 2) * chunk_bytes,
                         BLOCK, CT, B_SIMS);
        __builtin_amdgcn_s_wait_tensorcnt(1);  // chunk k+1 resident
      } else {
        __builtin_amdgcn_s_wait_tensorcnt(0);  // drain tail
      }
    }
    __syncthreads();
  }
}

extern "C" void kernel_launch(void* const* d_in, const int* in_sizes, int n_in,
                              void* d_out, int out_size, void* d_ws, size_t ws_size,
                              hipStream_t stream) {
  (void)in_sizes; (void)n_in; (void)out_size; (void)d_ws; (void)ws_size;
  const float* noise = (const float*)d_in[0];
  const float* sigma = (const float*)d_in[1];
  const float* mu    = (const float*)d_in[2];
  const float* lamb  = (const float*)d_in[3];
  const float* beta  = (const float*)d_in[4];
  const float* psi   = (const float*)d_in[5];
  const float* phi   = (const float*)d_in[6];
  const float* chi   = (const float*)d_in[7];
  float* out = (float*)d_out;

  dim3 grid(B_SIMS / BLOCK);   // 64 blocks
  dim3 block(BLOCK);           // 256 threads = 8 wave32
  balloon_windkessel_kernel<<<grid, block, 0, stream>>>(
      noise, sigma, mu, lamb, beta, psi, phi, chi, out);
}